// GLM_grating_81758997447012
// MI455X (gfx1250) — compile-verified
//
#include <hip/hip_runtime.h>
#include <hip/hip_bf16.h>
#include <math.h>

typedef __attribute__((ext_vector_type(2))) float v2f;
typedef __attribute__((ext_vector_type(8))) float v8f;

#define BATCH  128
#define NRN    54
#define DTH    4
#define STEPS  30
#define PIX    5625      // 75*75
#define KPAD   5632      // K padded to multiple of 64
#define NPAD   64
#define CELLS  (BATCH * NRN)   // 6912

// ---------------------------------------------------------------------------
// Kernel 1: 8x8 average pool  [128,600,600] -> xp_pad [128][KPAD], zero-padded.
// Bandwidth-bound (184 MB in); coalesced float4 loads, each thread = 1 output.
// ---------------------------------------------------------------------------
__global__ void pool_kernel(const float* __restrict__ x, float* __restrict__ xp) {
  int g = blockIdx.x * blockDim.x + threadIdx.x;
  if (g >= BATCH * KPAD) return;
  int b = g / KPAD;
  int p = g - b * KPAD;
  if (p >= PIX) { xp[g] = 0.0f; return; }      // zero K padding
  int pr = p / 75;
  int pc = p - pr * 75;
  const float* base = x + (size_t)b * 360000 + pr * 4800 + pc * 8;  // 16B aligned
  float s = 0.0f;
#pragma unroll
  for (int r = 0; r < 8; ++r) {
    const float4* q = reinterpret_cast<const float4*>(base + r * 600);
    float4 a0 = q[0], a1 = q[1];
    s += (a0.x + a0.y + a0.z + a0.w) + (a1.x + a1.y + a1.z + a1.w);
  }
  xp[g] = s * (1.0f / 64.0f);
}

// ---------------------------------------------------------------------------
// Kernel 2: transpose+pad fc_w [54][5625] -> wT [KPAD][NPAD] (zeros elsewhere).
// Coalesced writes; reads served from L2 (1.2 MB).
// ---------------------------------------------------------------------------
__global__ void wtrans_kernel(const float* __restrict__ w, float* __restrict__ wT) {
  int g = blockIdx.x * blockDim.x + threadIdx.x;
  if (g >= KPAD * NPAD) return;
  int k = g >> 6;
  int n = g & 63;
  wT[g] = (k < PIX && n < NRN) ? w[n * PIX + k] : 0.0f;
}

// ---------------------------------------------------------------------------
// Kernel 3: xf = xp @ wT + bias   via fp32 WMMA 16x16x4.
// grid = 8 blocks (M tiles), block = 128 threads = 4 waves, wave w -> N tile w.
// K chunks of 64 staged in LDS. sA row stride 68 floats => conflict-free
// 16-lane column reads (272 B rows keep float4 alignment).
// A frag (16x4): lanes 0-15: K={kk,kk+1}; lanes 16-31: K={kk+2,kk+3} (ISA 7.12.2)
// B frag (4x16): VGPR v holds rows K=v (lanes 0-15) and K=v+2 (lanes 16-31).
// ---------------------------------------------------------------------------
__global__ void __launch_bounds__(128) fc_wmma_kernel(const float* __restrict__ xp,
                                                      const float* __restrict__ wT,
                                                      const float* __restrict__ bias,
                                                      float* __restrict__ xf) {
  __shared__ float sA[16 * 68];
  __shared__ float sB[64 * NPAD];
  const int tid  = threadIdx.x;
  const int lane = tid & 31;
  const int wave = tid >> 5;          // N tile id (0..3)
  const int m0   = blockIdx.x * 16;
  const int half = lane >> 4;
  const int l15  = lane & 15;

  v8f acc = {0.f, 0.f, 0.f, 0.f, 0.f, 0.f, 0.f, 0.f};

  for (int k0 = 0; k0 < KPAD; k0 += 64) {
    // stage A: 16 rows x 64 K, float4 coalesced
    for (int i = tid; i < 16 * 16; i += 128) {
      int m = i >> 4, q = (i & 15) << 2;
      float4 a = *reinterpret_cast<const float4*>(xp + (size_t)(m0 + m) * KPAD + k0 + q);
      *reinterpret_cast<float4*>(&sA[m * 68 + q]) = a;
    }
    // stage B: 64 K rows x 64 N, float4 coalesced
    for (int i = tid; i < 64 * 16; i += 128) {
      int k = i >> 4, q = (i & 15) << 2;
      float4 bv = *reinterpret_cast<const float4*>(wT + (size_t)(k0 + k) * NPAD + q);
      *reinterpret_cast<float4*>(&sB[k * NPAD + q]) = bv;
    }
    __syncthreads();

#pragma unroll 4
    for (int kk = 0; kk < 64; kk += 4) {
      int ka = kk + (half << 1);
      v2f a;
      a.x = sA[l15 * 68 + ka];
      a.y = sA[l15 * 68 + ka + 1];
      v2f bf;
      bf.x = sB[ka * NPAD + (wave << 4) + l15];
      bf.y = sB[(ka + 1) * NPAD + (wave << 4) + l15];
      acc = __builtin_amdgcn_wmma_f32_16x16x4_f32(
          /*neg_a=*/false, a, /*neg_b=*/false, bf,
          /*c_mod=*/(short)0, acc, /*reuse_a=*/false, /*reuse_b=*/false);
    }
    __syncthreads();
  }

  // D layout (16x16 f32): VGPR v -> M = v + 8*half, N = l15 (ISA 7.12.2)
#pragma unroll
  for (int v = 0; v < 8; ++v) {
    int m = m0 + v + (half << 3);
    int n = (wave << 4) + l15;
    float o = acc[v] + (n < NRN ? bias[n] : 0.0f);
    xf[m * NPAD + n] = o;
  }
}

// ---------------------------------------------------------------------------
// Kernel 4: 30-step spiking GLM scan. Serially dependent through a global
// mean each step -> single workgroup (1 WGP), 1024 threads (32 waves),
// all state (hist[4], count, xf, l_weight row) in VGPRs; per-batch lateral
// sums and global rate mean via LDS + wave32 shuffles. 4 barriers/step.
// ---------------------------------------------------------------------------
#define NTH  1024
#define MAXC 7            // ceil(6912/1024)
__global__ void __launch_bounds__(NTH) scan_kernel(const float* __restrict__ xf,
                                                   const float* __restrict__ lw,  // [NRN][DTH]
                                                   const float* __restrict__ hw,  // [DTH]
                                                   float* __restrict__ out) {
  __shared__ float sVals[CELLS];       // per-cell s_{b,n}
  __shared__ float sBatch[BATCH];      // per-batch sum S_b
  __shared__ float sWave[NTH / 32];
  __shared__ float sThr;

  const int tid  = threadIdx.x;
  const int lane = tid & 31;
  const int wv   = tid >> 5;

  float h0[MAXC], h1[MAXC], h2[MAXC], h3[MAXC], cnt[MAXC], xv[MAXC];
  float lw0[MAXC], lw1[MAXC], lw2[MAXC], lw3[MAXC];
  const float w0 = hw[0], w1 = hw[1], w2 = hw[2], w3 = hw[3];

#pragma unroll
  for (int j = 0; j < MAXC; ++j) {
    int c = tid + (j << 10);
    h0[j] = h1[j] = h2[j] = h3[j] = cnt[j] = 0.0f;
    if (c < CELLS) {
      int b = c / NRN, n = c - b * NRN;
      xv[j]  = xf[b * NPAD + n];
      lw0[j] = lw[n * DTH + 0]; lw1[j] = lw[n * DTH + 1];
      lw2[j] = lw[n * DTH + 2]; lw3[j] = lw[n * DTH + 3];
    } else {
      xv[j] = lw0[j] = lw1[j] = lw2[j] = lw3[j] = 0.0f;
    }
  }

  for (int step = 0; step < STEPS; ++step) {
    float sv[MAXC], hsv[MAXC], rate[MAXC];
    // s_{b,n} = <hist, l_weight[n]>,  hs = <hist, h_weight>
#pragma unroll
    for (int j = 0; j < MAXC; ++j) {
      int c = tid + (j << 10);
      sv[j]  = h0[j] * lw0[j] + h1[j] * lw1[j] + h2[j] * lw2[j] + h3[j] * lw3[j];
      hsv[j] = h0[j] * w0 + h1[j] * w1 + h2[j] * w2 + h3[j] * w3;
      if (c < CELLS) sVals[c] = sv[j];
    }
    __syncthreads();
    // S_b = sum over the batch's 54 neurons
    if (tid < BATCH) {
      float acc = 0.0f;
      const float* row = &sVals[tid * NRN];
#pragma unroll
      for (int n = 0; n < NRN; ++n) acc += row[n];
      sBatch[tid] = acc;
    }
    __syncthreads();
    // rate = exp(xf + (S_b - s) + hs); accumulate global sum
    float psum = 0.0f;
#pragma unroll
    for (int j = 0; j < MAXC; ++j) {
      int c = tid + (j << 10);
      if (c < CELLS) {
        int b = c / NRN;
        float cur = xv[j] + (sBatch[b] - sv[j]) + hsv[j];
        rate[j] = __expf(cur);
        psum += rate[j];
      } else {
        rate[j] = 0.0f;
      }
    }
#pragma unroll
    for (int off = 16; off > 0; off >>= 1) psum += __shfl_down(psum, off, 32);
    if (lane == 0) sWave[wv] = psum;
    __syncthreads();
    if (tid < 32) {
      float v = sWave[tid];
#pragma unroll
      for (int off = 16; off > 0; off >>= 1) v += __shfl_down(v, off, 32);
      if (tid == 0) sThr = v * (1.0f / (float)CELLS);
    }
    __syncthreads();
    const float thr = sThr;
    // spike = rate > mean(rate); count += spike; shift history
#pragma unroll
    for (int j = 0; j < MAXC; ++j) {
      float sp = (rate[j] > thr) ? 1.0f : 0.0f;
      cnt[j] += sp;
      h0[j] = h1[j]; h1[j] = h2[j]; h2[j] = h3[j]; h3[j] = sp;
    }
  }

  // softplus(count), numerically stable
#pragma unroll
  for (int j = 0; j < MAXC; ++j) {
    int c = tid + (j << 10);
    if (c < CELLS) {
      float cc = cnt[j];
      out[c] = fmaxf(cc, 0.0f) + log1pf(__expf(-fabsf(cc)));
    }
  }
}

// ---------------------------------------------------------------------------
extern "C" void kernel_launch(void* const* d_in, const int* in_sizes, int n_in,
                              void* d_out, int out_size, void* d_ws, size_t ws_size,
                              hipStream_t stream) {
  const float* x  = (const float*)d_in[0];   // [128,1,600,600]
  const float* lw = (const float*)d_in[1];   // [54,4]
  const float* hw = (const float*)d_in[2];   // [4]
  const float* fw = (const float*)d_in[3];   // [54,5625]
  const float* fb = (const float*)d_in[4];   // [54]

  // workspace layout (~4.4 MB): xp_pad | wT | xf
  float* ws = (float*)d_ws;
  float* xp = ws;                                    // [128][KPAD]
  float* wT = xp + (size_t)BATCH * KPAD;             // [KPAD][NPAD]
  float* xf = wT + (size_t)KPAD * NPAD;              // [128][NPAD]
  float* out = (float*)d_out;                        // [128][54]

  pool_kernel<<<(BATCH * KPAD + 255) / 256, 256, 0, stream>>>(x, xp);
  wtrans_kernel<<<(KPAD * NPAD + 255) / 256, 256, 0, stream>>>(fw, wT);
  fc_wmma_kernel<<<BATCH / 16, 128, 0, stream>>>(xp, wT, fb, xf);
  scan_kernel<<<1, NTH, 0, stream>>>(xf, lw, hw, out);
}